// LASLNNet_56891136802926
// MI455X (gfx1250) — compile-verified
//
#include <hip/hip_runtime.h>

typedef __attribute__((ext_vector_type(2))) float v2f;
typedef __attribute__((ext_vector_type(8))) float v8f;

#define SAMP 16
#define XS_STRIDE 2401           // global x: 7^4 per sample (unpadded)
#define XS2_STRIDE 2748          // padded LDS x: 7*7*7*8 = 2744, +4 bank pad (even, /4)
#define XL 392                   // padded x strides: l=7*7*8
#define XD 56                    //                  d=7*8
#define XH 8                     //                  h=8 (W row padded 7->8)
#define H1S 1500                 // padded h1 sample stride: 2ch * 750
#define H1C 750                  // h1 channel stride: 5*5*5*6
#define HL 150                   // h1 strides: l=5*5*6
#define HD 30                    //             d=5*6
#define HH 6                     //             h=6 (W row padded 5->6)
#define K1T 216                  // layer1 K: 27 triples * 8 -> 54 steps, exact
#define K2T 324                  // layer2 K: 2ch * 27 triples * 6 -> 81 steps, exact
#define T1 (K1T/4)               // 54
#define T2 (K2T/4)               // 81

__global__ __launch_bounds__(128, 1)
void conv4d_fused_wmma(const float* __restrict__ xg,
                       const float* __restrict__ w1,
                       const float* __restrict__ b1,
                       const float* __restrict__ w2,
                       const float* __restrict__ b2,
                       float* __restrict__ outg)
{
    __shared__ alignas(16) float xs2[SAMP * XS2_STRIDE];   // 175,872 B
    __shared__ alignas(16) float h1s[SAMP * H1S];          //  96,000 B
    __shared__ alignas(16) float bwp1[(K1T/2) * 32];       //  13,824 B pair-interleaved Toeplitz B1
    __shared__ alignas(16) float bwp2[(K2T/2) * 32];       //  20,736 B pair-interleaved Toeplitz B2
    __shared__ float bb1[16];
    __shared__ float bb2[16];
    __shared__ int   offA1[K1T];                           // k' -> float offset in padded x cube
    __shared__ int   offA2[K2T];                           // k' -> float offset in padded h1

    const int tid  = threadIdx.x;
    const int wave = tid >> 5;
    const int lane = tid & 31;
    const int n16  = lane & 15;     // A row (sample) / B,D column
    const int hi   = lane >> 4;     // K sub-block (+0 or +2)

    // ---- stage x: repack 7-float W rows into 8-float padded rows (pad slot = 0) ----
    {
        const float* gbase = xg + (size_t)blockIdx.x * SAMP * XS_STRIDE;
        for (int i = tid; i < SAMP * 343; i += 128) {     // 343 rows per sample
            int s = i / 343, r = i % 343;
            const float* g = gbase + s * XS_STRIDE + r * 7;
            float* d = xs2 + s * XS2_STRIDE + r * 8;      // 16B-aligned by construction
            float4 lo = { g[0], g[1], g[2], g[3] };
            float4 hi4 = { g[4], g[5], g[6], 0.0f };
            *(float4*)(d)     = lo;
            *(float4*)(d + 4) = hi4;
        }
    }
    // ---- zero h1 (pad slots must be finite: garbage * 0-weight must stay 0) ----
    {
        float4 z = {0.f, 0.f, 0.f, 0.f};
        for (int i = tid; i < (SAMP * H1S) / 4; i += 128) ((float4*)h1s)[i] = z;
    }
    // ---- layer1 tables: k' = kk*8 + w ----
    for (int k = tid; k < K1T; k += 128) {
        int kk = k >> 3, w = k & 7;
        offA1[k] = (kk/9)*XL + ((kk/3)%3)*XD + (kk%3)*XH + w;
    }
    for (int i = tid; i < (K1T/2)*32; i += 128) {          // bwp1[pair][n][h]
        int j = i >> 5, r = i & 31, n = r >> 1, h = r & 1;
        int k = 2*j + h, kk = k >> 3, w = k & 7;
        float v = 0.0f;
        if (n < 10) {
            int o = n / 5, pw = n % 5, rr = w - pw;
            if (rr >= 0 && rr < 3) v = w1[o*81 + kk*3 + rr];
        }
        bwp1[i] = v;
    }
    // ---- layer2 tables: k' = (c*27 + kk)*6 + w ----
    for (int k = tid; k < K2T; k += 128) {
        int g = k / 6, w = k % 6, c = g / 27, kk = g % 27;
        offA2[k] = c*H1C + (kk/9)*HL + ((kk/3)%3)*HD + (kk%3)*HH + w;
    }
    for (int i = tid; i < (K2T/2)*32; i += 128) {          // bwp2[pair][n][h]
        int j = i >> 5, r = i & 31, n = r >> 1, h = r & 1;
        int k = 2*j + h, g = k / 6, w = k % 6, c = g / 27, kk = g % 27;
        float v = 0.0f;
        if (n < 12) {
            int o = n / 3, pw = n % 3, rr = w - pw;
            if (rr >= 0 && rr < 3) v = w2[o*162 + c*81 + kk*3 + rr];
        }
        bwp2[i] = v;
    }
    if (tid < 16) {
        bb1[tid] = (tid < 10) ? b1[tid / 5] : 0.0f;
        bb2[tid] = (tid < 12) ? b2[tid / 3] : 0.0f;
    }
    __syncthreads();

    // ======== layer 1: D[16 samp, 2ch x 5pw], 54 steps, 125 (L,D,H)-triples ========
    v2f bf1[T1];                 // hoisted B fragments (one b64 each, done once)
    int pk1[T1/2];               // hoisted pair-base offsets, two 16-bit per VGPR
    #pragma unroll
    for (int t = 0; t < T1; ++t) {
        int k0 = 4*t + 2*hi;
        bf1[t] = *(const v2f*)&bwp1[(k0 >> 1)*32 + n16*2];
    }
    #pragma unroll
    for (int j = 0; j < T1/2; ++j)
        pk1[j] = offA1[8*j + 2*hi] | (offA1[8*j + 4 + 2*hi] << 16);

    const int   aBase1 = n16 * XS2_STRIDE;
    const float bias1  = bb1[n16];
    const int   oSt1   = n16 / 5, pwSt1 = n16 % 5;

    for (int q = wave; q < 125; q += 4) {                  // wave-uniform -> EXEC all-ones
        int pl = q / 25, pd = (q / 5) % 5, ph = q % 5;
        int abase = aBase1 + pl*XL + pd*XD + ph*XH;
        v8f acc = {0.f,0.f,0.f,0.f,0.f,0.f,0.f,0.f};
        #pragma unroll
        for (int t = 0; t < T1; ++t) {
            int o = (t & 1) ? (pk1[t >> 1] >> 16) : (pk1[t >> 1] & 0xffff);
            v2f a = *(const v2f*)&xs2[abase + o];           // one ds_load_b64
            acc = __builtin_amdgcn_wmma_f32_16x16x4_f32(false, a, false, bf1[t],
                                                        (short)0, acc, false, false);
        }
        if (n16 < 10) {
            int pbase = pl*HL + pd*HD + ph*HH;              // padded h1 position base
            #pragma unroll
            for (int r = 0; r < 8; ++r) {
                int s = r + 8*hi;
                float v = acc[r] + bias1;
                h1s[s*H1S + oSt1*H1C + pbase + pwSt1] = fmaxf(v, 0.0f);
            }
        }
    }
    __syncthreads();

    // ======== layer 2: D[16 samp, 4ch x 3pw], 81 steps, 27 (L,D,H)-triples ========
    int pk2[(T2 + 1) / 2];       // 41 packed pair-base offsets
    #pragma unroll
    for (int j = 0; j < (T2 + 1) / 2; ++j) {
        int lo = offA2[8*j + 2*hi];
        int hw = (2*j + 1 < T2) ? offA2[8*j + 4 + 2*hi] : 0;
        pk2[j] = lo | (hw << 16);
    }
    const int   aBase2 = n16 * H1S;
    const float bias2  = bb2[n16];
    const int   oSt2   = n16 / 3, pwSt2 = n16 % 3;
    float* outb = outg + (size_t)blockIdx.x * SAMP * 324;

    for (int q = wave; q < 27; q += 4) {
        int pl = q / 9, pd = (q / 3) % 3, ph = q % 3;
        int abase = aBase2 + pl*HL + pd*HD + ph*HH;
        v8f acc = {0.f,0.f,0.f,0.f,0.f,0.f,0.f,0.f};
        #pragma unroll
        for (int t = 0; t < T2; ++t) {
            int k0 = 4*t + 2*hi;
            int o = (t & 1) ? (pk2[t >> 1] >> 16) : (pk2[t >> 1] & 0xffff);
            v2f a = *(const v2f*)&h1s[abase + o];            // ds_load_b64
            v2f b = *(const v2f*)&bwp2[(k0 >> 1)*32 + n16*2];// ds_load_b64, imm offset
            acc = __builtin_amdgcn_wmma_f32_16x16x4_f32(false, a, false, b,
                                                        (short)0, acc, false, false);
        }
        if (n16 < 12) {
            int pbase = q * 3;                               // true output p = q*3 + pw
            #pragma unroll
            for (int r = 0; r < 8; ++r) {
                int s = r + 8*hi;
                float v = acc[r] + bias2;
                outb[(size_t)s*324 + oSt2*81 + pbase + pwSt2] = fmaxf(v, 0.0f);
            }
        }
    }
}

extern "C" void kernel_launch(void* const* d_in, const int* in_sizes, int n_in,
                              void* d_out, int out_size, void* d_ws, size_t ws_size,
                              hipStream_t stream) {
    const float* x  = (const float*)d_in[0];
    const float* w1 = (const float*)d_in[1];
    const float* b1 = (const float*)d_in[2];
    const float* w2 = (const float*)d_in[3];
    const float* b2 = (const float*)d_in[4];
    float* out = (float*)d_out;

    int B    = in_sizes[0] / XS_STRIDE;   // 16384
    int nblk = B / SAMP;                  // 1024 workgroups of 16 samples
    hipLaunchKernelGGL(conv4d_fused_wmma, dim3(nblk), dim3(128), 0, stream,
                       x, w1, b1, w2, b2, out);
}